// GAT_54726473285976
// MI455X (gfx1250) — compile-verified
//
#include <hip/hip_runtime.h>
#include <hip/hip_bf16.h>
#include <stdint.h>

// ---------------------------------------------------------------------------
// GAT forward for MI455X (gfx1250, wave32).
//   N=50000, E=800000, IN=128, HID=32, H=4, OUT=64, ED=8, D=H*HID=128
//   * Node GEMMs: V_WMMA_F32_16X16X4_F32, K fully unrolled (32 WMMAs/wave),
//     compile-time shapes -> immediate-offset global loads, no u64 mul math.
//   * ea@We collapsed: a_e[e,h] = ea[e,:8] . (We @ a_edge[h])  (8xH precompute)
//   * 3-pass scatter-softmax, templated on <H,C> so all /H,%H,/C are shifts.
//   * LayerNorm+ELU: one wave32 per node, shfl_xor reductions.
// ---------------------------------------------------------------------------

typedef __attribute__((ext_vector_type(2))) float v2f;
typedef __attribute__((ext_vector_type(8))) float v8f;

#define GN  50000
#define GE  800000LL
#define GE2 (GE + 2LL * GN)   // 900000 after two rounds of self loops
#define GED 8

// ---------------- small utility kernels ----------------

__global__ void k_fill(float* __restrict__ p, float v, size_t n) {
  size_t i = (size_t)blockIdx.x * blockDim.x + threadIdx.x;
  size_t s = (size_t)gridDim.x * blockDim.x;
  for (; i < n; i += s) p[i] = v;
}

// order-preserving float atomic max (init to -inf)
__device__ __forceinline__ void atomicMaxFloat(float* addr, float val) {
  if (val >= 0.0f)
    atomicMax((int*)addr, __float_as_int(val));
  else
    atomicMin((unsigned int*)addr, (unsigned int)__float_as_uint(val));
}

// deg[v] = (#edges with dst==v) + 1 (self loop);  sum_ea[v] = sum edge_attr over dst
__global__ void k_deg_sum(const long long* __restrict__ ei,
                          const float* __restrict__ ea,
                          float* __restrict__ deg, float* __restrict__ sum_ea,
                          long long E) {
  long long e = (long long)blockIdx.x * blockDim.x + threadIdx.x;
  if (e >= E) return;
  long long d = ei[E + e];   // edge_index[1, e]
  atomicAdd(&deg[d], 1.0f);
  #pragma unroll
  for (int j = 0; j < GED; ++j)
    atomicAdd(&sum_ea[d * GED + j], ea[e * GED + j]);
}

__global__ void k_mean_div(float* __restrict__ mean_ea, const float* __restrict__ deg,
                           size_t n) {
  size_t i = (size_t)blockIdx.x * blockDim.x + threadIdx.x;
  if (i >= n) return;
  mean_ea[i] = mean_ea[i] / deg[i / GED];   // deg >= 1 always
}

// Wea[h][d] = sum_c We[d, h*C+c] * a_edge[h, c]   (tiny: H*8 <= 32 threads)
__global__ void k_wea(const float* __restrict__ We, const float* __restrict__ a_edge,
                      float* __restrict__ wea, int H, int C) {
  int idx = threadIdx.x;
  if (idx >= H * GED) return;
  int h = idx / GED, d = idx % GED;
  float s = 0.0f;
  for (int c = 0; c < C; ++c)
    s += We[d * (H * C) + h * C + c] * a_edge[h * C + c];
  wea[h * GED + d] = s;
}

// ---------------- WMMA GEMM: C[M,Nc] = A[M,K] @ B[K,Nc] (+ bias) ----------------
// Compile-time K and NT (N tiles). grid.x = M/16, blockDim.x = NT*32.
// One wave per 16x16 output tile, K fully unrolled -> 32 WMMAs back-to-back.
template <int K, int NT>
__global__ void k_gemm_wmma_f32(const float* __restrict__ A, const float* __restrict__ B,
                                const float* __restrict__ bias, float* __restrict__ C) {
  constexpr int Nc = NT * 16;
  const int wave = threadIdx.x >> 5;
  const int lane = threadIdx.x & 31;
  const int half = lane >> 4;       // 0: lanes 0-15, 1: lanes 16-31
  const int l16  = lane & 15;
  const int mBase = blockIdx.x * 16;
  const int nBase = wave * 16;

  // Per-lane constant bases: all loop loads become immediate offsets.
  const float* __restrict__ ap = A + (size_t)(mBase + l16) * K + 2 * half;
  const float* __restrict__ bp = B + (size_t)(2 * half) * Nc + (nBase + l16);

  __builtin_prefetch(ap, 0, 1);           // global_prefetch_b8 (A row: 512B)
  __builtin_prefetch(ap + 64, 0, 1);

  v8f acc = {};
  #pragma unroll
  for (int k0 = 0; k0 < K; k0 += 4) {
    // A frag (16x4 f32): VGPR0 = K{0|2}, VGPR1 = K{1|3} split by half-wave
    v2f a = *(const v2f*)(ap + k0);                 // 8B-aligned vector load
    // B frag (4x16 f32): mirrored K striping, row striped across lanes
    v2f b;
    b.x = bp[k0 * Nc];
    b.y = bp[(k0 + 1) * Nc];
    acc = __builtin_amdgcn_wmma_f32_16x16x4_f32(false, a, false, b,
                                                (short)0, acc, false, false);
  }
  // C/D layout: VGPR i -> M = i (+8 for upper half-wave), N = lane%16
  const int orow0 = mBase + half * 8;
  const int ocol  = nBase + l16;
  const float bv = bias ? bias[ocol] : 0.0f;
  #pragma unroll
  for (int i = 0; i < 8; ++i)
    C[(size_t)(orow0 + i) * Nc + ocol] = acc[i] + bv;
}

// ---------------- attention coefficients: as/ad[n,h] = h[n,h,:] . a_{src,dst}[h]
template <int H, int C>
__global__ void k_attn_coeff(const float* __restrict__ hf,
                             const float* __restrict__ a_src, const float* __restrict__ a_dst,
                             float* __restrict__ as, float* __restrict__ ad, int N) {
  int idx = blockIdx.x * blockDim.x + threadIdx.x;
  if (idx >= N * H) return;
  int n = idx / H, h = idx % H;
  const float* row = hf + (size_t)n * H * C + h * C;
  float s = 0.0f, d = 0.0f;
  #pragma unroll 4
  for (int c = 0; c < C; ++c) {
    float v = row[c];
    s += v * a_src[h * C + c];
    d += v * a_dst[h * C + c];
  }
  as[idx] = s;
  ad[idx] = d;
}

// edge src/dst/ea resolution for the augmented edge list
__device__ __forceinline__ void edge_sd(const long long* ei, long long e,
                                        long long& s, long long& d, int& ea_kind,
                                        long long& v) {
  if (e < GE)            { s = ei[e]; d = ei[GE + e]; ea_kind = 0; v = e; }
  else if (e < GE + GN)  { v = e - GE;       s = d = v; ea_kind = 1; }     // zero attrs
  else                   { v = e - GE - GN;  s = d = v; ea_kind = 2; }     // mean attrs
}

// ---------------- edge pass 1: logits + scatter max ----------------
template <int H>
__global__ void k_edge_logits(const long long* __restrict__ ei,
                              const float* __restrict__ ea,
                              const float* __restrict__ mean_ea,
                              const float* __restrict__ as, const float* __restrict__ ad,
                              const float* __restrict__ wea,
                              float* __restrict__ wlog, float* __restrict__ mmax) {
  long long idx = (long long)blockIdx.x * blockDim.x + threadIdx.x;
  if (idx >= GE2 * H) return;
  long long e = idx / H; int h = (int)(idx % H);
  long long s, d, v; int kind;
  edge_sd(ei, e, s, d, kind, v);
  float ae = 0.0f;
  if (kind != 1) {
    const float* eap = (kind == 0) ? (ea + v * GED) : (mean_ea + v * GED);
    #pragma unroll
    for (int j = 0; j < GED; ++j) ae += eap[j] * wea[h * GED + j];
  }
  float x = as[s * H + h] + ad[d * H + h] + ae;
  float lg = (x > 0.0f) ? x : 0.2f * x;          // leaky_relu, slope 0.2
  wlog[idx] = lg;
  atomicMaxFloat(&mmax[d * H + h], lg);
}

// ---------------- edge pass 2: w = exp(l - m[dst]); denom scatter sum ----------------
template <int H>
__global__ void k_edge_exp(const long long* __restrict__ ei,
                           float* __restrict__ wlog, const float* __restrict__ mmax,
                           float* __restrict__ denom) {
  long long idx = (long long)blockIdx.x * blockDim.x + threadIdx.x;
  if (idx >= GE2 * H) return;
  long long e = idx / H; int h = (int)(idx % H);
  long long s, d, v; int kind;
  edge_sd(ei, e, s, d, kind, v);
  float w = expf(wlog[idx] - mmax[d * H + h]);
  wlog[idx] = w;
  atomicAdd(&denom[d * H + h], w);
}

// ---------------- edge pass 3: agg[dst] += alpha * h[src] (one wave32 per edge) ----
// HC/32 consecutive features per lane; C % PER == 0 => one head per lane block.
template <int H, int C>
__global__ void k_edge_msg(const long long* __restrict__ ei,
                           const float* __restrict__ wlog, const float* __restrict__ denom,
                           const float* __restrict__ hf, float* __restrict__ agg) {
  constexpr int HC = H * C;
  constexpr int PER = HC / 32;                // 4 (H=4,C=32) or 2 (H=1,C=64)
  int lane = threadIdx.x & 31;
  long long e = (long long)blockIdx.x * (blockDim.x >> 5) + (threadIdx.x >> 5);
  if (e >= GE2) return;
  long long s, d, v; int kind;
  edge_sd(ei, e, s, d, kind, v);
  const int f = lane * PER;
  const int h = f / C;                        // constant shifts after templating
  float alpha = wlog[e * H + h] / (denom[d * H + h] + 1e-16f);
  const float* __restrict__ src = hf + s * (long long)HC + f;
  float* __restrict__ dstp = agg + d * (long long)HC + f;
  float hv[PER];
  #pragma unroll
  for (int i = 0; i < PER; ++i) hv[i] = src[i];   // vectorizes to b128/b64 gather
  #pragma unroll
  for (int i = 0; i < PER; ++i) atomicAdd(&dstp[i], alpha * hv[i]);
}

// ---------------- LayerNorm(agg+bias) -> ELU -> (+identity) ; D=128, wave per node --
__global__ void k_ln_elu(const float* __restrict__ agg, const float* __restrict__ bias,
                         const float* __restrict__ g, const float* __restrict__ b,
                         const float* __restrict__ identity, float* __restrict__ out,
                         int N) {
  const int D = 128;
  int lane = threadIdx.x & 31;
  int n = blockIdx.x * (blockDim.x >> 5) + (threadIdx.x >> 5);
  if (n >= N) return;
  float val[4];
  float sum = 0.0f;
  #pragma unroll
  for (int i = 0; i < 4; ++i) {
    int k = lane * 4 + i;
    val[i] = agg[(size_t)n * D + k] + bias[k];
    sum += val[i];
  }
  #pragma unroll
  for (int m = 16; m >= 1; m >>= 1) sum += __shfl_xor(sum, m, 32);
  float mu = sum * (1.0f / D);
  float var = 0.0f;
  #pragma unroll
  for (int i = 0; i < 4; ++i) { float t = val[i] - mu; var += t * t; }
  #pragma unroll
  for (int m = 16; m >= 1; m >>= 1) var += __shfl_xor(var, m, 32);
  var *= (1.0f / D);
  float rs = rsqrtf(var + 1e-5f);
  #pragma unroll
  for (int i = 0; i < 4; ++i) {
    int k = lane * 4 + i;
    float y = (val[i] - mu) * rs * g[k] + b[k];
    y = (y > 0.0f) ? y : (expf(y) - 1.0f);       // ELU(alpha=1)
    if (identity) y += identity[(size_t)n * D + k];
    out[(size_t)n * D + k] = y;
  }
}

// ---------------- final output: out = agg + b  (H=1, mean over 1 head = identity) ---
__global__ void k_out_bias(const float* __restrict__ agg, const float* __restrict__ b,
                           float* __restrict__ out, int N, int D) {
  int idx = blockIdx.x * blockDim.x + threadIdx.x;
  if (idx >= N * D) return;
  out[idx] = agg[idx] + b[idx % D];
}

// ---------------------------------------------------------------------------

static inline unsigned cdiv(long long a, long long b) { return (unsigned)((a + b - 1) / b); }

struct ConvBufs {
  float *wea, *hfeat, *agg, *asrc, *adst, *mmax, *denom, *wlog, *mean_ea;
  const float* ea;
  const long long* ei;
};

template <int H, int C>
static void run_conv(const ConvBufs& B, const float* xin, const float* W,
                     const float* asv, const float* adv, const float* We,
                     const float* aev, hipStream_t stream) {
  constexpr int HC = H * C;
  const unsigned FB = 256;
  const float NEG_INF = -__builtin_huge_valf();
  k_wea<<<1, 32, 0, stream>>>(We, aev, B.wea, H, C);
  k_gemm_wmma_f32<128, HC / 16><<<GN / 16, (HC / 16) * 32, 0, stream>>>(xin, W, nullptr,
                                                                        B.hfeat);
  k_attn_coeff<H, C><<<cdiv((long long)GN * H, FB), FB, 0, stream>>>(B.hfeat, asv, adv,
                                                                     B.asrc, B.adst, GN);
  k_fill<<<1024, FB, 0, stream>>>(B.mmax, NEG_INF, (size_t)GN * H);
  k_fill<<<1024, FB, 0, stream>>>(B.denom, 0.0f, (size_t)GN * H);
  k_fill<<<4096, FB, 0, stream>>>(B.agg, 0.0f, (size_t)GN * HC);
  k_edge_logits<H><<<cdiv(GE2 * H, FB), FB, 0, stream>>>(B.ei, B.ea, B.mean_ea, B.asrc,
                                                         B.adst, B.wea, B.wlog, B.mmax);
  k_edge_exp<H><<<cdiv(GE2 * H, FB), FB, 0, stream>>>(B.ei, B.wlog, B.mmax, B.denom);
  k_edge_msg<H, C><<<cdiv(GE2, FB / 32), FB, 0, stream>>>(B.ei, B.wlog, B.denom, B.hfeat,
                                                          B.agg);
}

extern "C" void kernel_launch(void* const* d_in, const int* in_sizes, int n_in,
                              void* d_out, int out_size, void* d_ws, size_t ws_size,
                              hipStream_t stream) {
  if (n_in < 27) return;
  // setup_inputs() insertion order: x, edge_attr, params{conv1,conv2,convf,ln0,ln1,skip0}, edge_index
  const float* x    = (const float*)d_in[0];
  const float* ea   = (const float*)d_in[1];
  const float* W1   = (const float*)d_in[2];
  const float* as1  = (const float*)d_in[3];
  const float* ad1  = (const float*)d_in[4];
  const float* We1  = (const float*)d_in[5];
  const float* ae1  = (const float*)d_in[6];
  const float* b1   = (const float*)d_in[7];
  const float* W2   = (const float*)d_in[8];
  const float* as2  = (const float*)d_in[9];
  const float* ad2  = (const float*)d_in[10];
  const float* We2  = (const float*)d_in[11];
  const float* ae2  = (const float*)d_in[12];
  const float* b2   = (const float*)d_in[13];
  const float* Wf   = (const float*)d_in[14];
  const float* asf  = (const float*)d_in[15];
  const float* adf  = (const float*)d_in[16];
  const float* Wef  = (const float*)d_in[17];
  const float* aef  = (const float*)d_in[18];
  const float* bf   = (const float*)d_in[19];
  const float* ln0g = (const float*)d_in[20];
  const float* ln0b = (const float*)d_in[21];
  const float* ln1g = (const float*)d_in[22];
  const float* ln1b = (const float*)d_in[23];
  const float* Ws   = (const float*)d_in[24];
  const float* bs   = (const float*)d_in[25];
  const long long* ei = (const long long*)d_in[26];
  float* out = (float*)d_out;

  // workspace layout (floats)
  float* ws = (float*)d_ws;
  size_t off = 0;
  auto take = [&](size_t n) { float* p = ws + off; off += n; return p; };
  float* deg     = take(GN);                 // augmented in-degree (+self loop)
  float* mean_ea = take((size_t)GN * GED);   // scatter-mean edge attrs
  float* wea     = take(4 * GED);            // per-conv collapsed We.a_edge
  float* hfeat   = take((size_t)GN * 128);   // transformed features of current conv
  float* agg     = take((size_t)GN * 128);   // scatter-sum accumulator
  float* asrc    = take((size_t)GN * 4);
  float* adst    = take((size_t)GN * 4);
  float* mmax    = take((size_t)GN * 4);
  float* denom   = take((size_t)GN * 4);
  float* wlog    = take((size_t)GE2 * 4);    // logits -> softmax weights
  float* hprev   = take((size_t)GN * 128);   // activations between layers
  float* ident   = take((size_t)GN * 128);   // skip connection
  (void)ws_size;

  const unsigned FB = 256;

  // ---- edge augmentation statistics (deg starts at 1 for the self loop)
  k_fill<<<2048, FB, 0, stream>>>(deg, 1.0f, GN);
  k_fill<<<2048, FB, 0, stream>>>(mean_ea, 0.0f, (size_t)GN * GED);
  k_deg_sum<<<cdiv(GE, FB), FB, 0, stream>>>(ei, ea, deg, mean_ea, GE);
  k_mean_div<<<cdiv((long long)GN * GED, FB), FB, 0, stream>>>(mean_ea, deg,
                                                               (size_t)GN * GED);

  ConvBufs B{wea, hfeat, agg, asrc, adst, mmax, denom, wlog, mean_ea, ea, ei};

  // ---- conv1 (128 -> 4x32) + LN0 + ELU
  run_conv<4, 32>(B, x, W1, as1, ad1, We1, ae1, stream);
  k_ln_elu<<<cdiv(GN, FB / 32), FB, 0, stream>>>(agg, b1, ln0g, ln0b, nullptr, hprev, GN);

  // ---- skip projection: identity = hprev @ Ws + bs
  k_gemm_wmma_f32<128, 8><<<GN / 16, 256, 0, stream>>>(hprev, Ws, bs, ident);

  // ---- conv2 (128 -> 4x32) + LN1 + ELU + residual
  run_conv<4, 32>(B, hprev, W2, as2, ad2, We2, ae2, stream);
  k_ln_elu<<<cdiv(GN, FB / 32), FB, 0, stream>>>(agg, b2, ln1g, ln1b, ident, hprev, GN);

  // ---- convf (128 -> 1x64), concat=False with 1 head => + bias only
  run_conv<1, 64>(B, hprev, Wf, asf, adf, Wef, aef, stream);
  k_out_bias<<<cdiv((long long)GN * 64, FB), FB, 0, stream>>>(agg, bf, out, GN, 64);
}